// PostProcess_28123445854384
// MI455X (gfx1250) — compile-verified
//
#include <hip/hip_runtime.h>
#include <hip/hip_bf16.h>
#include <stdint.h>

#define BB     16
#define NN     25200
#define NC     80
#define ROW    85            // xy(2)+wh(2)+obj(1)+cls(80)
#define KTOP   4096
#define CAP    8192          // candidate capacity per image (expected ~2.4K)
#define MAXDET 300
#define CONF_T 0.45f
#define IOU_T  0.45f
#define MAX_WH 4096.0f

typedef int v4i __attribute__((vector_size(16)));
typedef __attribute__((address_space(1))) v4i* gv4i_p;   // global int4*
typedef __attribute__((address_space(3))) v4i* lv4i_p;   // LDS int4*

// ---------------------------------------------------------------------------
// Kernel 0: zero the per-image candidate counters
// ---------------------------------------------------------------------------
__global__ void k_init(unsigned int* __restrict__ counters) {
  if (threadIdx.x < BB) counters[threadIdx.x] = 0u;
}

// ---------------------------------------------------------------------------
// Kernel 1: streaming filter. One thread per box. Early-out on obj<=CONF
// (~95% of threads), then per-class score test and atomic append of a
// sort key that reproduces lax.top_k ordering:
//   key = (bits(score) << 32) | (0xFFFFFFFF - flat_idx)
// score > 0.45 > 0 so float bits are order-preserving.
// ---------------------------------------------------------------------------
__global__ __launch_bounds__(256)
void k_filter(const float* __restrict__ pred,
              unsigned int* __restrict__ counters,
              unsigned long long* __restrict__ keys) {
  int gid = blockIdx.x * blockDim.x + threadIdx.x;
  if (gid >= BB * NN) return;
  int b = gid / NN;
  int n = gid - b * NN;
  const float* row = pred + (size_t)gid * ROW;
  // gfx1250 prefetch path (lowered to global_prefetch_b8)
  __builtin_prefetch(row + (size_t)ROW * 2048, 0, 1);
  float obj = row[4];
  if (!(obj > CONF_T)) return;
  #pragma unroll 4
  for (int c = 0; c < NC; ++c) {
    float s = __fmul_rn(row[5 + c], obj);
    if (s > CONF_T) {
      unsigned int slot = atomicAdd(&counters[b], 1u);
      if (slot < CAP) {
        unsigned int idx = (unsigned int)n * NC + (unsigned int)c;
        unsigned long long key =
            ((unsigned long long)__float_as_uint(s) << 32) |
            (unsigned long long)(0xFFFFFFFFu - idx);
        keys[(size_t)b * CAP + slot] = key;
      }
    }
  }
}

// ---------------------------------------------------------------------------
// Kernel 2: per-image bitonic sort of 8192 u64 keys in LDS (descending),
// then decode the top-4096 into raw xyxy boxes, class-offset boxes,
// scores and class ids.
// ---------------------------------------------------------------------------
__global__ __launch_bounds__(1024)
void k_sort(const float* __restrict__ pred,
            const unsigned int* __restrict__ counters,
            const unsigned long long* __restrict__ keys,
            float4* __restrict__ rawbox,
            float4* __restrict__ offbox,
            float* __restrict__ score,
            float* __restrict__ clsf) {
  __shared__ unsigned long long sk[CAP];   // 64 KB
  int b = blockIdx.x;
  unsigned int c = counters[b];
  if (c > CAP) c = CAP;
  for (int t = threadIdx.x; t < CAP; t += blockDim.x)
    sk[t] = (t < (int)c) ? keys[(size_t)b * CAP + t] : 0ull;
  __syncthreads();

  // bitonic sort, descending
  for (unsigned int k = 2; k <= CAP; k <<= 1) {
    for (unsigned int j = k >> 1; j > 0; j >>= 1) {
      for (unsigned int i = threadIdx.x; i < CAP; i += blockDim.x) {
        unsigned int ixj = i ^ j;
        if (ixj > i) {
          unsigned long long a = sk[i], d = sk[ixj];
          bool up = ((i & k) == 0);
          if (up ? (a < d) : (a > d)) { sk[i] = d; sk[ixj] = a; }
        }
      }
      __syncthreads();
    }
  }

  for (int t = threadIdx.x; t < KTOP; t += blockDim.x) {
    unsigned long long key = sk[t];
    size_t o = (size_t)b * KTOP + t;
    if (key == 0ull) {                       // padding: invalid entry
      rawbox[o] = make_float4(0.f, 0.f, 0.f, 0.f);
      offbox[o] = make_float4(0.f, 0.f, 0.f, 0.f);
      score[o]  = -1.0f;
      clsf[o]   = 0.0f;
      continue;
    }
    float s = __uint_as_float((unsigned int)(key >> 32));
    unsigned int idx = 0xFFFFFFFFu - (unsigned int)(key & 0xFFFFFFFFull);
    unsigned int bi = idx / NC;
    unsigned int ci = idx - bi * NC;
    const float* row = pred + ((size_t)b * NN + bi) * ROW;
    float x = row[0], y = row[1], w = row[2], h = row[3];
    float hw = __fmul_rn(w, 0.5f), hh = __fmul_rn(h, 0.5f);
    float x1 = __fsub_rn(x, hw), y1 = __fsub_rn(y, hh);
    float x2 = __fadd_rn(x, hw), y2 = __fadd_rn(y, hh);
    float cf = (float)ci;
    float off = __fmul_rn(cf, MAX_WH);
    rawbox[o] = make_float4(x1, y1, x2, y2);
    offbox[o] = make_float4(__fadd_rn(x1, off), __fadd_rn(y1, off),
                            __fadd_rn(x2, off), __fadd_rn(y2, off));
    score[o] = s;
    clsf[o]  = cf;
  }
}

// ---------------------------------------------------------------------------
// Kernel 3: per-image greedy NMS over the sorted 4096, then compact the
// kept rows into dets[300][6] + mask[300]. Boxes staged into LDS via the
// gfx1250 async global->LDS path (GLOBAL_LOAD_ASYNC_TO_LDS_B128).
// ---------------------------------------------------------------------------
__global__ __launch_bounds__(1024)
void k_nms(const float4* __restrict__ offbox,
           const float4* __restrict__ rawbox,
           const float* __restrict__ score,
           const float* __restrict__ clsf,
           float* __restrict__ dets,      // [B,300,6]
           float* __restrict__ mask) {    // [B,300] (bool as 0/1 float)
  __shared__ float4       sb[KTOP];          // 64 KB
  __shared__ unsigned int skeep[KTOP / 32];  // 512 B
  __shared__ unsigned int spref[KTOP / 32];
  __shared__ unsigned int stotal;
  int b = blockIdx.x;
  const float4* src = offbox + (size_t)b * KTOP;

#if __has_builtin(__builtin_amdgcn_global_load_async_to_lds_b128)
  for (int t = threadIdx.x; t < KTOP; t += blockDim.x) {
    __builtin_amdgcn_global_load_async_to_lds_b128(
        (gv4i_p)(src + t), (lv4i_p)(&sb[t]), 0, 0);
  }
  #if __has_builtin(__builtin_amdgcn_s_wait_asynccnt)
  __builtin_amdgcn_s_wait_asynccnt(0);
  #else
  asm volatile("s_wait_asynccnt 0" ::: "memory");
  #endif
#else
  for (int t = threadIdx.x; t < KTOP; t += blockDim.x) sb[t] = src[t];
#endif

  // keep-bit init: keep = (score > CONF)
  for (int w = threadIdx.x; w < KTOP / 32; w += blockDim.x) skeep[w] = 0u;
  // zero this image's dets region while we're here
  for (int t = threadIdx.x; t < MAXDET * 6; t += blockDim.x)
    dets[(size_t)b * MAXDET * 6 + t] = 0.0f;
  __syncthreads();
  for (int t = threadIdx.x; t < KTOP; t += blockDim.x)
    if (score[(size_t)b * KTOP + t] > CONF_T)
      atomicOr(&skeep[t >> 5], 1u << (t & 31));
  __syncthreads();

  // greedy NMS scan (identical to the reference lax.scan)
  for (int i = 0; i < KTOP; ++i) {
    unsigned int wi = skeep[i >> 5];              // uniform read
    if (!((wi >> (i & 31)) & 1u)) continue;       // uniform branch
    float4 bi4 = sb[i];
    float area_i = __fmul_rn(__fsub_rn(bi4.z, bi4.x), __fsub_rn(bi4.w, bi4.y));
    for (int j = i + 1 + (int)threadIdx.x; j < KTOP; j += (int)blockDim.x) {
      if (!((skeep[j >> 5] >> (j & 31)) & 1u)) continue;
      float4 bj = sb[j];
      float lx = fmaxf(bi4.x, bj.x), ly = fmaxf(bi4.y, bj.y);
      float rx = fminf(bi4.z, bj.z), ry = fminf(bi4.w, bj.w);
      float iw = fmaxf(__fsub_rn(rx, lx), 0.0f);
      float ih = fmaxf(__fsub_rn(ry, ly), 0.0f);
      float inter  = __fmul_rn(iw, ih);
      float area_j = __fmul_rn(__fsub_rn(bj.z, bj.x), __fsub_rn(bj.w, bj.y));
      float denom  = __fadd_rn(__fsub_rn(__fadd_rn(area_i, area_j), inter), 1e-7f);
      float iou    = __fdiv_rn(inter, denom);
      if (iou > IOU_T) atomicAnd(&skeep[j >> 5], ~(1u << (j & 31)));
    }
    __syncthreads();
  }

  // prefix scan of keep bits -> ranks
  if (threadIdx.x == 0) {
    unsigned int run = 0;
    for (int w = 0; w < KTOP / 32; ++w) { spref[w] = run; run += __popc(skeep[w]); }
    stotal = run;
  }
  __syncthreads();

  unsigned int nk = stotal; if (nk > MAXDET) nk = MAXDET;
  for (int t = threadIdx.x; t < MAXDET; t += blockDim.x)
    mask[(size_t)b * MAXDET + t] = (t < (int)nk) ? 1.0f : 0.0f;

  for (int t = threadIdx.x; t < KTOP; t += blockDim.x) {
    unsigned int wrd = skeep[t >> 5];
    if (!((wrd >> (t & 31)) & 1u)) continue;
    unsigned int rank = spref[t >> 5] + __popc(wrd & ((1u << (t & 31)) - 1u));
    if (rank < MAXDET) {
      size_t o = (size_t)b * KTOP + t;
      float4 rb = rawbox[o];
      float* dr = dets + ((size_t)b * MAXDET + rank) * 6;
      dr[0] = rb.x; dr[1] = rb.y; dr[2] = rb.z; dr[3] = rb.w;
      dr[4] = score[o]; dr[5] = clsf[o];
    }
  }
}

// ---------------------------------------------------------------------------
extern "C" void kernel_launch(void* const* d_in, const int* in_sizes, int n_in,
                              void* d_out, int out_size, void* d_ws, size_t ws_size,
                              hipStream_t stream) {
  (void)in_sizes; (void)n_in; (void)out_size;
  const float* pred = (const float*)d_in[0];

  // workspace layout (256B aligned blocks)
  char* ws = (char*)d_ws;
  size_t off = 0;
  auto take = [&](size_t bytes) { char* p = ws + off; off = (off + bytes + 255) & ~(size_t)255; return p; };
  unsigned int*       counters = (unsigned int*)      take(BB * sizeof(unsigned int));
  unsigned long long* keys     = (unsigned long long*)take((size_t)BB * CAP * sizeof(unsigned long long));
  float4*             rawbox   = (float4*)            take((size_t)BB * KTOP * sizeof(float4));
  float4*             offbox   = (float4*)            take((size_t)BB * KTOP * sizeof(float4));
  float*              score    = (float*)             take((size_t)BB * KTOP * sizeof(float));
  float*              clsf     = (float*)             take((size_t)BB * KTOP * sizeof(float));
  (void)ws_size;

  float* dets = (float*)d_out;                       // [B,300,6]
  float* mask = dets + (size_t)BB * MAXDET * 6;      // [B,300]

  k_init<<<1, 64, 0, stream>>>(counters);
  int total = BB * NN;
  k_filter<<<(total + 255) / 256, 256, 0, stream>>>(pred, counters, keys);
  k_sort<<<BB, 1024, 0, stream>>>(pred, counters, keys, rawbox, offbox, score, clsf);
  k_nms<<<BB, 1024, 0, stream>>>(offbox, rawbox, score, clsf, dets, mask);
}